// DeepFM_54391465836775
// MI455X (gfx1250) — compile-verified
//
#include <hip/hip_runtime.h>
#include <hip/hip_bf16.h>

// ---------------------------------------------------------------------------
// DeepFM forward for MI455X (gfx1250, wave32).
//
// Roofline: sparse_onehot is 852 MB and must be fully scanned -> ~37us at
// 23.3 TB/s HBM; streamed with NT loads so it doesn't thrash the 192MB L2.
// All GEMMs together are ~1 GFLOP fp32 -> done with V_WMMA_F32_16X16X4_F32,
// weights staged per-N-tile into LDS so the WMMA inner loop is DS-only
// (no global-latency stalls; 256 waves for latency hiding).
// ---------------------------------------------------------------------------

typedef float  v2f __attribute__((ext_vector_type(2)));
typedef float  v4f __attribute__((ext_vector_type(4)));
typedef float  v8f __attribute__((ext_vector_type(8)));

#define BATCH      4096
#define NDENSE     13
#define NSPARSE    26
#define FEATN      2000
#define EMB        8
#define DIM        221          // 13 + 26*8
#define DPAD       224          // DIM padded to multiple of 4 (WMMA K step)
#define KFM        10
#define HIDN       200
#define NPAD       208          // 200 padded to multiple of 16
#define TILE_ROWS  128          // rows per workgroup in kernel 2
#define S0         228          // LDS stride, 224-col buffer (bank-friendly)
#define S1         212          // LDS stride, 208-col buffer (bank-friendly)
#define KP         228          // LDS stride of transposed weight panel

static __device__ __forceinline__ v8f wmma_f32_16x16x4(v2f a, v2f b, v8f c) {
    // D = A(16x4) * B(4x16) + C(16x16), fp32. 8-arg VOP3P form:
    // (neg_a, A, neg_b, B, c_mod, C, reuse_a, reuse_b)
    return __builtin_amdgcn_wmma_f32_16x16x4_f32(false, a, false, b,
                                                 (short)0, c, false, false);
}

// ---------------------------------------------------------------------------
// Kernel 0: fill dense columns and zero the pad columns of stack[B, 224].
// Embedding columns 13..220 are written by kernel 1.
// ---------------------------------------------------------------------------
__global__ __launch_bounds__(256)
void k0_fill_dense(const float* __restrict__ dense, float* __restrict__ stack) {
    int t = blockIdx.x * blockDim.x + threadIdx.x;   // covers BATCH*DPAD exactly
    int b = t / DPAD;
    int c = t - b * DPAD;
    if (c < NDENSE) {
        stack[(size_t)b * DPAD + c] = dense[b * NDENSE + c];
    } else if (c >= DIM) {
        stack[(size_t)b * DPAD + c] = 0.0f;
    }
}

// ---------------------------------------------------------------------------
// Kernel 1: one wave per (b, f) pair. Streams 2000 floats of one-hot row as
// NT float4 (512B/wave/iter, coalesced, no L2 pollution), skips all-zero
// vectors (common case: 1 nonzero in 2000), accumulates v*table[f,n,0..7],
// butterfly-reduces across the wave, lane 0 writes the 8 embedding floats.
// This kernel carries ~all of the HBM traffic (852 MB).
// ---------------------------------------------------------------------------
__global__ __launch_bounds__(256)
void k1_embed(const float* __restrict__ onehot,
              const float* __restrict__ tables,
              float* __restrict__ stack) {
    const int lane = threadIdx.x & 31;
    const int pair = blockIdx.x * 8 + (threadIdx.x >> 5); // = b*26 + f
    const int b = pair / NSPARSE;
    const int f = pair - b * NSPARSE;

    const v4f* __restrict__ src =
        (const v4f*)(onehot + (size_t)pair * FEATN);      // 500 float4 per row

    float acc[EMB];
#pragma unroll
    for (int k = 0; k < EMB; ++k) acc[k] = 0.0f;

    for (int i = lane; i < FEATN / 4; i += 32) {
        v4f v = __builtin_nontemporal_load(src + i);      // TH=NT stream
        if (v.x != 0.0f || v.y != 0.0f || v.z != 0.0f || v.w != 0.0f) {
            const float vv[4] = {v.x, v.y, v.z, v.w};
#pragma unroll
            for (int c = 0; c < 4; ++c) {
                if (vv[c] != 0.0f) {
                    const float* tr = tables + ((size_t)f * FEATN + (4 * i + c)) * EMB;
#pragma unroll
                    for (int k = 0; k < EMB; ++k) acc[k] += vv[c] * tr[k];
                }
            }
        }
    }
    // wave32 butterfly reduction
#pragma unroll
    for (int off = 16; off > 0; off >>= 1) {
#pragma unroll
        for (int k = 0; k < EMB; ++k)
            acc[k] += __shfl_down(acc[k], off, 32);
    }
    if (lane == 0) {
        float* dst = stack + (size_t)b * DPAD + NDENSE + f * EMB;
#pragma unroll
        for (int k = 0; k < EMB; ++k) dst[k] = acc[k];
    }
}

// ---------------------------------------------------------------------------
// Kernel 2: fused FM wide part + 3-layer MLP + combine + sigmoid.
// Block = 256 threads (8 waves), tile = 128 batch rows, each wave owns one
// 16-row M-tile. Per N-tile, the 224x16 weight panel is cooperatively staged
// (transposed) into LDS, so the WMMA k-loop is ds_load_b64 + v_wmma only.
// ---------------------------------------------------------------------------
static __device__ __forceinline__
void layer_wmma(const float* __restrict__ lds_in, int in_stride,
                float* __restrict__ lds_out, int out_stride,
                const float* __restrict__ Wg, int Krows,
                const float* __restrict__ bg,
                int ksteps, float* __restrict__ panel,
                int wrb, int lane, int tid) {
    const int s      = lane >> 4;     // 0: K pair {k0,k0+1}, 1: {k0+2,k0+3}
    const int n_lane = lane & 15;     // B/C/D column within tile; A row M
    const int kmax   = ksteps * 4;

    for (int nt = 0; nt < NPAD / 16; ++nt) {
        __syncthreads();              // prev consumers of panel / prev layer done
        // ---- stage transposed, zero-padded weight panel: panel[n][k] ----
        for (int idx = tid; idx < 16 * kmax; idx += 256) {
            const int k  = idx >> 4;
            const int nl = idx & 15;                 // consecutive tid -> coalesced n
            const int col = nt * 16 + nl;
            float w = (col < HIDN && k < Krows) ? Wg[(size_t)k * HIDN + col] : 0.0f;
            panel[nl * KP + k] = w;
        }
        __syncthreads();

        // ---- DS-only WMMA loop ----
        const int  col  = nt * 16 + n_lane;
        const float bval = (col < HIDN) ? bg[col] : 0.0f;
        v8f acc;
#pragma unroll
        for (int g = 0; g < 8; ++g) acc[g] = bval;

        const float* arow = lds_in + (wrb + n_lane) * in_stride;
        const float* brow = panel + n_lane * KP;
#pragma unroll 4
        for (int ks = 0; ks < ksteps; ++ks) {
            const int k0 = ks * 4 + 2 * s;
            v2f a = *(const v2f*)(arow + k0);        // A[m][k0], A[m][k0+1]
            v2f b = *(const v2f*)(brow + k0);        // B[k0][n], B[k0+1][n]
            acc = wmma_f32_16x16x4(a, b, acc);
        }
        // ReLU + store. C/D layout: VGPR g -> row g (+8 for lanes 16-31).
        const int r0 = wrb + 8 * s;
#pragma unroll
        for (int g = 0; g < 8; ++g)
            lds_out[(r0 + g) * out_stride + col] = fmaxf(acc[g], 0.0f);
    }
}

__global__ __launch_bounds__(256)
void k2_fm_mlp(const float* __restrict__ stackg,
               const float* __restrict__ fm_w,  const float* __restrict__ fm_V,
               const float* __restrict__ W1,    const float* __restrict__ b1,
               const float* __restrict__ W2,    const float* __restrict__ b2,
               const float* __restrict__ W3,    const float* __restrict__ b3,
               const float* __restrict__ Wout,  const float* __restrict__ bout,
               const float* __restrict__ w1s,   const float* __restrict__ w2s,
               const float* __restrict__ biass, float* __restrict__ out) {
    extern __shared__ float smem[];
    float* buf0   = smem;                         // TILE_ROWS x S0
    float* buf1   = buf0 + TILE_ROWS * S0;        // TILE_ROWS x S1
    float* wide_l = buf1 + TILE_ROWS * S1;        // TILE_ROWS
    float* panel  = wide_l + TILE_ROWS;           // 16 x KP

    const int tid  = threadIdx.x;                 // 0..255
    const int lane = tid & 31;
    const int wrb  = (tid >> 5) * 16;             // wave's 16-row M-tile base
    const int base = blockIdx.x * TILE_ROWS;

    // ---- Stage A: stage stack tile into LDS (stride 228, zero pad cols) ----
    for (int idx = tid; idx < TILE_ROWS * (DPAD / 4); idx += 256) {
        int r  = idx / (DPAD / 4);
        int c4 = (idx - r * (DPAD / 4)) * 4;
        float4 v = *(const float4*)&stackg[(size_t)(base + r) * DPAD + c4];
        *(float4*)&buf0[r * S0 + c4] = v;
    }
    if (tid < TILE_ROWS) {
        float4 z = {0.0f, 0.0f, 0.0f, 0.0f};
        *(float4*)&buf0[tid * S0 + DPAD] = z;     // cols 224..227
    }
    __syncthreads();

    // ---- Stage B: FM wide part (1 row / thread; fm_* loads are uniform) ----
    if (tid < TILE_ROWS) {
        const int r = tid;
        const float* x = &buf0[r * S0];
        float first = 0.0f;
        float xv[KFM], x2v2[KFM];
#pragma unroll
        for (int j = 0; j < KFM; ++j) { xv[j] = 0.0f; x2v2[j] = 0.0f; }
        for (int k = 0; k < DIM; ++k) {
            const float xk  = x[k];
            const float xk2 = xk * xk;
            first += xk * fm_w[k];
            const float* Vr = fm_V + k * KFM;
#pragma unroll
            for (int j = 0; j < KFM; ++j) {
                const float v = Vr[j];
                xv[j]   += xk  * v;
                x2v2[j] += xk2 * v * v;
            }
        }
        float sec = 0.0f;
#pragma unroll
        for (int j = 0; j < KFM; ++j) sec += xv[j] * xv[j] - x2v2[j];
        wide_l[r] = first + 0.5f * sec;
    }
    // (no barrier needed: layer_wmma starts with __syncthreads)

    // ---- Stage C: 3 WMMA layers (LDS ping-pong, LDS-staged weights) ----
    layer_wmma(buf0, S0, buf1, S1, W1, DIM,  b1, DPAD / 4, panel, wrb, lane, tid);
    layer_wmma(buf1, S1, buf0, S1, W2, HIDN, b2, NPAD / 4, panel, wrb, lane, tid);
    layer_wmma(buf0, S1, buf1, S1, W3, HIDN, b3, NPAD / 4, panel, wrb, lane, tid);
    __syncthreads();

    // ---- Stage D: deep head + combine + sigmoid ----
    if (tid < TILE_ROWS) {
        const int r = tid;
        const float w1v = w1s[0], w2v = w2s[0], bv = biass[0], bo = bout[0];
        const float* h = &buf1[r * S1];
        float acc = 0.0f;
        for (int k = 0; k < HIDN; ++k) acc += h[k] * Wout[k];
        const float comb = w1v * wide_l[r] + w2v * (acc + bo) + bv;
        out[base + r] = 1.0f / (1.0f + expf(-comb));
    }
}

// ---------------------------------------------------------------------------
extern "C" void kernel_launch(void* const* d_in, const int* in_sizes, int n_in,
                              void* d_out, int out_size, void* d_ws, size_t ws_size,
                              hipStream_t stream) {
    const float* dense   = (const float*)d_in[0];
    const float* onehot  = (const float*)d_in[1];
    const float* tables  = (const float*)d_in[2];
    const float* fm_w    = (const float*)d_in[3];
    const float* fm_V    = (const float*)d_in[4];
    const float* W1      = (const float*)d_in[5];
    const float* b1      = (const float*)d_in[6];
    const float* W2      = (const float*)d_in[7];
    const float* b2      = (const float*)d_in[8];
    const float* W3      = (const float*)d_in[9];
    const float* b3      = (const float*)d_in[10];
    const float* Wout    = (const float*)d_in[11];
    const float* bout    = (const float*)d_in[12];
    const float* w1s     = (const float*)d_in[13];
    const float* w2s     = (const float*)d_in[14];
    const float* biass   = (const float*)d_in[15];
    float* out = (float*)d_out;

    float* stack = (float*)d_ws;                       // BATCH x DPAD fp32

    // k0: dense + pad columns of stack
    k0_fill_dense<<<(BATCH * DPAD) / 256, 256, 0, stream>>>(dense, stack);

    // k1: one-hot embedding scan (one wave per (b,f) pair; 8 waves/block)
    k1_embed<<<(BATCH * NSPARSE) / 8, 256, 0, stream>>>(onehot, tables, stack);

    // k2: fused FM + MLP with WMMA; 128 rows per block, dynamic LDS (~240KB)
    const size_t shmem = (size_t)(TILE_ROWS * S0 + TILE_ROWS * S1 + TILE_ROWS
                                  + 16 * KP) * sizeof(float);
    k2_fm_mlp<<<BATCH / TILE_ROWS, 256, shmem, stream>>>(
        stack, fm_w, fm_V, W1, b1, W2, b2, W3, b3, Wout, bout, w1s, w2s, biass, out);
}